// TopKRouter_6236292514568
// MI455X (gfx1250) — compile-verified
//
#include <hip/hip_runtime.h>
#include <math.h>

// Problem constants (from reference)
#define NUM_EXPERTS 64
#define HIDDEN      4096
#define NUM_K       2
#define CAPK        (256 * NUM_K)   // EXPERT_CAPACITY * NUM_K = 512
#define BATCH       8
#define SEQ         4096
#define T_TOTAL     (BATCH * SEQ)   // 32768 tokens

// Tiling
#define KCHUNK        64
#define TOK_PER_BLOCK 128           // 8 waves * 16 tokens
#define A_STRIDE      68            // padded LDS row stride (floats); 68*4=272, 16B aligned
#define W_STRIDE      68

typedef float v2f __attribute__((ext_vector_type(2)));
typedef float v8f __attribute__((ext_vector_type(8)));

// ---------------------------------------------------------------------------
// Kernel 1: router logits GEMM (fp32 WMMA) + softmax + top-2 per token.
// Memory-bound: streams 512MB of hidden states once; W (1MB) stays in L2.
// ---------------------------------------------------------------------------
__global__ __launch_bounds__(256) void router_gemm_softmax_topk(
    const float* __restrict__ hidden,   // [B*S, HIDDEN]
    const float* __restrict__ Wmat,     // [E, HIDDEN]
    const float* __restrict__ bias,     // [E]
    float* __restrict__ out_idx_list,   // [B*S, 2]  (indices as float)
    float* __restrict__ out_probs_topk, // [B*S, 2]
    float* __restrict__ out_logits,     // [B*S, E]
    int*   __restrict__ ws_top2)        // [B*S] packed: e1 | (e2<<8)
{
    __shared__ float lA[TOK_PER_BLOCK * A_STRIDE];  // 34816 B
    __shared__ float lW[NUM_EXPERTS   * W_STRIDE];  // 17408 B

    const int tid  = threadIdx.x;
    const int wave = tid >> 5;
    const int lane = tid & 31;
    const int half = lane >> 4;     // 0: K{0,1}, 1: K{2,3} per fp32 A/B layout
    const int l16  = lane & 15;     // A: row M; B: column N; C/D: column N
    const int tokBase = blockIdx.x * TOK_PER_BLOCK;

    v8f c0 = {}, c1 = {}, c2 = {}, c3 = {};   // N-tiles: experts 0-15,16-31,32-47,48-63

    for (int k0 = 0; k0 < HIDDEN; k0 += KCHUNK) {
        // ---- stage A tile: 128 tokens x 64 k (coalesced float4) ----
        #pragma unroll
        for (int i = 0; i < 8; ++i) {
            int idx = tid + i * 256;
            int row = idx >> 4;
            int c4  = (idx & 15) << 2;
            float4 v = *(const float4*)(hidden + (size_t)(tokBase + row) * HIDDEN + k0 + c4);
            *(float4*)&lA[row * A_STRIDE + c4] = v;
        }
        // ---- stage W tile: 64 experts x 64 k ----
        #pragma unroll
        for (int i = 0; i < 4; ++i) {
            int idx = tid + i * 256;
            int row = idx >> 4;
            int c4  = (idx & 15) << 2;
            float4 v = *(const float4*)(Wmat + (size_t)row * HIDDEN + k0 + c4);
            *(float4*)&lW[row * W_STRIDE + c4] = v;
        }
        __syncthreads();

        // ---- WMMA: 16 K-steps of 4, 4 N-tiles each ----
        const float* arow = &lA[(wave * 16 + l16) * A_STRIDE];
        const float* b0r  = &lW[(l16      ) * W_STRIDE];
        const float* b1r  = &lW[(16 + l16 ) * W_STRIDE];
        const float* b2r  = &lW[(32 + l16 ) * W_STRIDE];
        const float* b3r  = &lW[(48 + l16 ) * W_STRIDE];
        #pragma unroll 4
        for (int kk = 0; kk < KCHUNK; kk += 4) {
            const int ka = kk + 2 * half;
            v2f a;  a.x = arow[ka];  a.y = arow[ka + 1];
            v2f b0; b0.x = b0r[ka];  b0.y = b0r[ka + 1];
            v2f b1; b1.x = b1r[ka];  b1.y = b1r[ka + 1];
            v2f b2; b2.x = b2r[ka];  b2.y = b2r[ka + 1];
            v2f b3; b3.x = b3r[ka];  b3.y = b3r[ka + 1];
            c0 = __builtin_amdgcn_wmma_f32_16x16x4_f32(false, a, false, b0, (short)0, c0, false, false);
            c1 = __builtin_amdgcn_wmma_f32_16x16x4_f32(false, a, false, b1, (short)0, c1, false, false);
            c2 = __builtin_amdgcn_wmma_f32_16x16x4_f32(false, a, false, b2, (short)0, c2, false, false);
            c3 = __builtin_amdgcn_wmma_f32_16x16x4_f32(false, a, false, b3, (short)0, c3, false, false);
        }
        __syncthreads();
    }

    // ---- bias add (per lane, expert = tile*16 + l16 is constant across VGPRs) ----
    const float be0 = bias[l16], be1 = bias[16 + l16], be2 = bias[32 + l16], be3 = bias[48 + l16];
    #pragma unroll
    for (int v = 0; v < 8; ++v) { c0[v] += be0; c1[v] += be1; c2[v] += be2; c3[v] += be3; }

    // ---- scatter logits to LDS per C-layout: VGPR v, half -> token M = v + 8*half ----
    #pragma unroll
    for (int v = 0; v < 8; ++v) {
        int m = wave * 16 + v + 8 * half;
        lA[m * A_STRIDE + l16     ] = c0[v];
        lA[m * A_STRIDE + 16 + l16] = c1[v];
        lA[m * A_STRIDE + 32 + l16] = c2[v];
        lA[m * A_STRIDE + 48 + l16] = c3[v];
    }
    __syncthreads();

    // ---- coalesced logits store ----
    #pragma unroll
    for (int i = 0; i < 8; ++i) {
        int idx = tid + i * 256;
        int row = idx >> 4;
        int c4  = (idx & 15) << 2;
        float4 v;
        v.x = lA[row * A_STRIDE + c4 + 0];
        v.y = lA[row * A_STRIDE + c4 + 1];
        v.z = lA[row * A_STRIDE + c4 + 2];
        v.w = lA[row * A_STRIDE + c4 + 3];
        *(float4*)(out_logits + (size_t)(tokBase + row) * NUM_EXPERTS + c4) = v;
    }

    // ---- softmax + top-2 : one thread per token ----
    if (tid < TOK_PER_BLOCK) {
        const float* lr = &lA[tid * A_STRIDE];
        float mx = lr[0];
        #pragma unroll
        for (int e = 1; e < NUM_EXPERTS; ++e) mx = fmaxf(mx, lr[e]);
        float v1 = -3.4e38f, v2 = -3.4e38f, sum = 0.0f;
        int i1 = 0, i2 = 0;
        #pragma unroll
        for (int e = 0; e < NUM_EXPERTS; ++e) {
            float x = lr[e];
            sum += __expf(x - mx);
            if (x > v1)      { v2 = v1; i2 = i1; v1 = x; i1 = e; }
            else if (x > v2) { v2 = x; i2 = e; }
        }
        const int tok = tokBase + tid;
        const float inv = 1.0f / sum;
        out_idx_list[tok * 2 + 0]   = (float)i1;
        out_idx_list[tok * 2 + 1]   = (float)i2;
        out_probs_topk[tok * 2 + 0] = __expf(v1 - mx) * inv;
        out_probs_topk[tok * 2 + 1] = __expf(v2 - mx) * inv;
        ws_top2[tok] = i1 | (i2 << 8);
    }
}

// ---------------------------------------------------------------------------
// Kernel 2: per-(batch, expert) inclusive cumsum over tokens + capacity mask.
// One block per (b,e); 16 chunks of 256 tokens, LDS Hillis-Steele scan.
// ---------------------------------------------------------------------------
__global__ __launch_bounds__(256) void router_cumsum_mask(
    const int* __restrict__ ws_top2,          // [B*S] packed indices
    float* __restrict__ out_expert_index,     // [B*S, E]
    float* __restrict__ out_mask)             // [B*S, E]
{
    __shared__ int buf[256];
    const int b   = blockIdx.x >> 6;
    const int e   = blockIdx.x & 63;
    const int tid = threadIdx.x;

    int base = 0;
    for (int chunk = 0; chunk < SEQ / 256; ++chunk) {
        const int s = chunk * 256 + tid;
        const int packed = ws_top2[b * SEQ + s];
        const int bit = (((packed & 255) == e) | (((packed >> 8) & 255) == e)) ? 1 : 0;
        buf[tid] = bit;
        __syncthreads();
        #pragma unroll
        for (int off = 1; off < 256; off <<= 1) {
            int t = (tid >= off) ? buf[tid - off] : 0;
            __syncthreads();
            buf[tid] += t;
            __syncthreads();
        }
        const int prio = base + buf[tid];   // inclusive cumsum (token_priority)
        base += buf[255];
        __syncthreads();                    // everyone done reading buf before next chunk writes

        const int m = (prio <= CAPK) ? 1 : 0;
        const size_t o = ((size_t)(b * SEQ + s)) * NUM_EXPERTS + e;
        out_mask[o]         = (float)m;
        out_expert_index[o] = (float)(bit & m);
    }
}

// ---------------------------------------------------------------------------
extern "C" void kernel_launch(void* const* d_in, const int* in_sizes, int n_in,
                              void* d_out, int out_size, void* d_ws, size_t ws_size,
                              hipStream_t stream) {
    const float* hidden = (const float*)d_in[0];   // [B,S,H] f32
    const float* Wmat   = (const float*)d_in[1];   // [E,H]   f32
    const float* bias   = (const float*)d_in[2];   // [E]     f32

    float* out = (float*)d_out;
    // Outputs concatenated flat in reference return order:
    float* out_idx_list     = out;                                   // [B,S,2]
    float* out_expert_index = out + (size_t)T_TOTAL * NUM_K;         // [B,S,E]
    float* out_mask         = out_expert_index + (size_t)T_TOTAL * NUM_EXPERTS;
    float* out_probs_topk   = out_mask + (size_t)T_TOTAL * NUM_EXPERTS;
    float* out_logits       = out_probs_topk + (size_t)T_TOTAL * NUM_K;

    int* ws_top2 = (int*)d_ws;   // [B*S] packed top-2 indices (128 KB)

    router_gemm_softmax_topk<<<T_TOTAL / TOK_PER_BLOCK, 256, 0, stream>>>(
        hidden, Wmat, bias, out_idx_list, out_probs_topk, out_logits, ws_top2);

    router_cumsum_mask<<<BATCH * NUM_EXPERTS, 256, 0, stream>>>(
        ws_top2, out_expert_index, out_mask);
}